// ResGraphConv_24163486007803
// MI455X (gfx1250) — compile-verified
//
#include <hip/hip_runtime.h>

// CDNA5 / gfx1250, wave32.
// fp32 WMMA 16x16x4 GEMMs with the weight tile staged into LDS by the
// Tensor Data Mover (tensor_load_to_lds + s_wait_tensorcnt), plus an
// atomic-scatter kernel for the graph aggregation.

typedef __attribute__((ext_vector_type(2))) float        v2f;
typedef __attribute__((ext_vector_type(4))) float        v4f;
typedef __attribute__((ext_vector_type(8))) float        v8f;
typedef __attribute__((ext_vector_type(4))) unsigned int v4u;
typedef __attribute__((ext_vector_type(4))) int          v4i;
typedef __attribute__((ext_vector_type(8))) int          v8i;

#define V_NODES 200000
#define E_EDGES 600000
#define D_IN    128
#define D_OUT   256

// ---------------------------------------------------------------------------
// TDM: DMA a 64-row x K-col f32 tile of W (row-major, row stride K) into LDS
// at byte offset 0, padding each K-dword row with 4 dwords so the LDS row
// stride is (K+4) floats (bank-conflict-free fragment reads).
// D# layout per CDNA5 ISA ch.8 (group0 128b, group1 256b; groups 2/3 zero).
// ---------------------------------------------------------------------------
template <int K>
__device__ __forceinline__ void tdm_stage_w_tile(const float* wtile) {
  const unsigned long long ga = (unsigned long long)wtile;

  v4u g0;
  g0[0] = 1u;                                           // count=1, user D#, no gather
  g0[1] = 0u;                                           // lds_addr = 0 (dyn LDS base)
  g0[2] = (unsigned)(ga & 0xFFFFFFFFu);                 // global_addr[31:0]
  g0[3] = (unsigned)((ga >> 32) & 0x01FFFFFFu)          // global_addr[56:32]
          | (2u << 30);                                 // type = 2 ("image")

  constexpr unsigned padInterval = (K == 128) ? 6u : 7u; // 128 / 256 dwords
  constexpr unsigned padAmount   = 3u;                   // 4 dwords of pad

  v8i g1;
  g1[0] = (int)((2u << 16)                 // data_size = 4 bytes
                | (1u << 20)               // pad_enable
                | (padInterval << 22)
                | (padAmount << 25));      // workgroup_mask = 0 (not in cluster)
  g1[1] = (int)(((unsigned)K & 0xFFFFu) << 16);          // tensor_dim0[15:0]
  g1[2] = (int)((((unsigned)K >> 16) & 0xFFFFu)          // tensor_dim0[31:16]
                | (64u << 16));                          // tensor_dim1[15:0] = 64 rows
  g1[3] = (int)(((unsigned)K) << 16);                    // tensor_dim1 hi = 0, tile_dim0 = K
  g1[4] = (int)(64u);                                    // tile_dim1 = 64, tile_dim2 = 0
  g1[5] = (int)K;                                        // tensor_dim0_stride = K
  g1[6] = 0;
  g1[7] = 0;

  const v4i z4 = {0, 0, 0, 0};                           // groups 2/3 unused (2-D)
  const v8i z8 = {0, 0, 0, 0, 0, 0, 0, 0};
  __builtin_amdgcn_tensor_load_to_lds(g0, g1, z4, z4, z8, 0);
}

// ---------------------------------------------------------------------------
// Y[v, n] = (ACCUM ? Y[v,n] : 0) + sum_k act(X[v,k]) * W[n,k] + bias[n]
// One wave computes a 16x64 output strip (4x 16x16 f32 WMMA tiles, A reused).
// 256-thread block = 8 waves = 8 consecutive row tiles; grid.y = 256/64.
// B tile lives in LDS (TDM-staged, stride K+4 floats).
// ---------------------------------------------------------------------------
template <int K, bool RELU, bool ACCUM>
__global__ __launch_bounds__(256) void gemm_wmma_f32(
    const float* __restrict__ X, const float* __restrict__ W,
    const float* __restrict__ bias, float* __restrict__ Y, int V) {
  constexpr int LS = K + 4;                 // LDS row stride in floats
  constexpr int NOUT = D_OUT;
  extern __shared__ float bW[];             // 64 * LS floats

  const unsigned lane = threadIdx.x & 31u;
  const unsigned wave = threadIdx.x >> 5;
  const unsigned lrow = lane & 15u;         // row-in-tile (A) / col-in-tile (B,D)
  const unsigned hsel = lane >> 4u;         // half select (K pair / row half)
  const int colBase = blockIdx.y * 64;

  // Wave 0 DMAs W[colBase..colBase+63][0..K-1] into LDS; EXEC-independent op.
  if (threadIdx.x < 32u) {
    tdm_stage_w_tile<K>(W + (size_t)colBase * K);
    __builtin_amdgcn_s_wait_tensorcnt(0);
  }
  __syncthreads();

  const int rowTile = blockIdx.x * 8 + (int)wave;   // wave-uniform
  if (rowTile * 16 >= V) return;                    // whole-wave exit, EXEC stays full
  const int rowBase = rowTile * 16;

  // A frag: lane holds X[rowBase+lrow, k0 + 2*hsel .. +1]
  const float* arow = X + (size_t)(rowBase + (int)lrow) * K + 2u * hsel;
  // B frag from LDS: lane holds W[colBase+lrow+16t, k0 + 2*hsel .. +1]
  const float* bl = bW + (size_t)lrow * LS + 2u * hsel;

  v8f acc[4];
  acc[0] = v8f{}; acc[1] = v8f{}; acc[2] = v8f{}; acc[3] = v8f{};

#pragma unroll 8
  for (int k0 = 0; k0 < K; k0 += 4) {
    v2f a = *(const v2f*)(arow + k0);
    if constexpr (RELU) {
      a[0] = fmaxf(a[0], 0.0f);
      a[1] = fmaxf(a[1], 0.0f);
    }
    v2f b0 = *(const v2f*)(bl + k0);
    v2f b1 = *(const v2f*)(bl + 16 * LS + k0);
    v2f b2 = *(const v2f*)(bl + 32 * LS + k0);
    v2f b3 = *(const v2f*)(bl + 48 * LS + k0);
    acc[0] = __builtin_amdgcn_wmma_f32_16x16x4_f32(false, a, false, b0,
                                                   (short)0, acc[0], false, false);
    acc[1] = __builtin_amdgcn_wmma_f32_16x16x4_f32(false, a, false, b1,
                                                   (short)0, acc[1], false, false);
    acc[2] = __builtin_amdgcn_wmma_f32_16x16x4_f32(false, a, false, b2,
                                                   (short)0, acc[2], false, false);
    acc[3] = __builtin_amdgcn_wmma_f32_16x16x4_f32(false, a, false, b3,
                                                   (short)0, acc[3], false, false);
  }

  // Epilogue: lane owns column col = colBase+16t+lrow; VGPR r holds row r+8*hsel.
#pragma unroll
  for (int t = 0; t < 4; ++t) {
    const int col = colBase + 16 * t + (int)lrow;
    const float bv = bias[col];
    float* ycol = Y + (size_t)(rowBase + 8 * (int)hsel) * NOUT + col;
#pragma unroll
    for (int r = 0; r < 8; ++r) {
      float v = acc[t][r] + bv;
      if constexpr (ACCUM) v += ycol[(size_t)r * NOUT];
      ycol[(size_t)r * NOUT] = v;
    }
  }
}

// ---------------------------------------------------------------------------
// For each edge (s,d): H[s,:] += Y[d,:]; H[d,:] += Y[s,:]   (rows of 256 f32)
// 64 threads per edge, 4 floats per thread; non-temporal 16B gathers (rows
// are consumed once) + global f32 atomics for the accumulation.
// ---------------------------------------------------------------------------
__global__ __launch_bounds__(256) void scatter_add_edges(
    const long long* __restrict__ edge, const float* __restrict__ Y,
    float* __restrict__ H, int E) {
  const int gid = blockIdx.x * blockDim.x + threadIdx.x;
  const int e = gid >> 6;
  if (e >= E) return;
  const int t = gid & 63;
  const long long s = edge[2 * (long long)e + 0];
  const long long d = edge[2 * (long long)e + 1];

  const size_t so = (size_t)s * D_OUT;
  const size_t dofs = (size_t)d * D_OUT;
  const int c = t * 4;

  const v4f yd = __builtin_nontemporal_load((const v4f*)(Y + dofs + c));
  const v4f ys = __builtin_nontemporal_load((const v4f*)(Y + so + c));

  float* hs = H + so + c;
  float* hd = H + dofs + c;
  atomicAdd(hs + 0, yd[0]); atomicAdd(hs + 1, yd[1]);
  atomicAdd(hs + 2, yd[2]); atomicAdd(hs + 3, yd[3]);
  atomicAdd(hd + 0, ys[0]); atomicAdd(hd + 1, ys[1]);
  atomicAdd(hd + 2, ys[2]); atomicAdd(hd + 3, ys[3]);
}

extern "C" void kernel_launch(void* const* d_in, const int* in_sizes, int n_in,
                              void* d_out, int out_size, void* d_ws, size_t ws_size,
                              hipStream_t stream) {
  const float*     x      = (const float*)d_in[0];
  const long long* edge   = (const long long*)d_in[1];   // int64 per reference
  const float*     w0_1   = (const float*)d_in[2];
  const float*     b0_1   = (const float*)d_in[3];
  const float*     w1_1   = (const float*)d_in[4];
  const float*     b1_1   = (const float*)d_in[5];
  const float*     w0_2   = (const float*)d_in[6];
  const float*     b0_2   = (const float*)d_in[7];
  const float*     w1_2   = (const float*)d_in[8];
  const float*     b1_2   = (const float*)d_in[9];
  const float*     w_skip = (const float*)d_in[10];
  const float*     b_skip = (const float*)d_in[11];
  float* out = (float*)d_out;

  float* h1 = (float*)d_ws;                       // V x 256
  float* y1 = h1 + (size_t)V_NODES * D_OUT;       // V x 256 (reused per layer)

  const dim3 blk(256);
  const dim3 gg((V_NODES / 16 + 7) / 8, D_OUT / 64);        // 1563 x 4
  const int scat_blocks = (E_EDGES * 64 + 255) / 256;       // 150000
  const size_t shm128 = (size_t)64 * (128 + 4) * sizeof(float);  // 33.8 KB
  const size_t shm256 = (size_t)64 * (256 + 4) * sizeof(float);  // 66.6 KB

  // Layer 1: h1 = relu(x)@w0_1^T + b0_1 ; y1 = relu(x)@w1_1^T + b1_1
  gemm_wmma_f32<128, true, false><<<gg, blk, shm128, stream>>>(x, w0_1, b0_1, h1, V_NODES);
  gemm_wmma_f32<128, true, false><<<gg, blk, shm128, stream>>>(x, w1_1, b1_1, y1, V_NODES);
  // h1 += segment sums of y1 over edges (both directions)
  scatter_add_edges<<<scat_blocks, blk, 0, stream>>>(edge, y1, h1, E_EDGES);

  // out = x@w_skip^T + b_skip         (skip path, no relu)
  gemm_wmma_f32<128, false, false><<<gg, blk, shm128, stream>>>(x, w_skip, b_skip, out, V_NODES);
  // out += relu(h1)@w0_2^T + b0_2     (layer-2 self term accumulated in place)
  gemm_wmma_f32<256, true, true><<<gg, blk, shm256, stream>>>(h1, w0_2, b0_2, out, V_NODES);
  // y1 = relu(h1)@w1_2^T + b1_2
  gemm_wmma_f32<256, true, false><<<gg, blk, shm256, stream>>>(h1, w1_2, b1_2, y1, V_NODES);
  // out += segment sums of y1 over edges
  scatter_add_edges<<<scat_blocks, blk, 0, stream>>>(edge, y1, out, E_EDGES);
}